// Model_21569325760600
// MI455X (gfx1250) — compile-verified
//
#include <hip/hip_runtime.h>
#include <hip/hip_bf16.h>

// ---------------------------------------------------------------------------
// Protein frame/atom construction for MI455X (gfx1250, wave32).
//
// Memory-bound (~140 MB traffic, ~0.75 GFLOP -> ~3 FLOP/B << any compute
// knee; HBM floor ~6us at 23.3 TB/s). WMMA does not apply (data-dependent
// 3x3 frame chains + gathers). CDNA5 paths used instead:
//   * GLOBAL_LOAD_ASYNC_TO_LDS_B128 / _B32 (ASYNCcnt) to stage the ~46 KB
//     (ss,restype) tables into LDS once per block.
//   * LDS as per-thread frame scratch so the rigids_dep atom gather is
//     3x ds_load_b128 instead of ~2000 v_cndmask selects per thread.
//   * Bank-padding (entry strides 100/76/25/9 dwords) to keep <=2-way
//     conflicts on the 64-bank LDS.
// ---------------------------------------------------------------------------

#define MAX_TORSION 7
#define MAX_RIGID   8
#define MAX_SS      3
#define MAX_ATOM    24
#define NRES        22
#define EPSF        1e-6f

#define BLK       96      // 3 waves; 300000 = 96 * 3125 exactly (no tail)
#define FR_STRIDE 100     // dwords per-thread frame scratch (8 frames * 12 + pad)
#define T_STRIDE  100     // padded dwords per (ss,rt) transforms entry (96 used)
#define G_STRIDE  76      // padded dwords per (ss,rt) rigids entry (72 used)
#define RD_STRIDE 25      // padded ints per restype rdep row (24 used)
#define TD_STRIDE 9       // padded ints per restype tdep row (8 used)

#define A_DW   (BLK * FR_STRIDE)          // 9600 dw union region: transforms then frames
#define B_DW   (MAX_SS * NRES * G_STRIDE) // 5016 dw rigids
#define DEP_RD 0
#define DEP_TD (NRES * RD_STRIDE)         // 550
#define DEP_N  (NRES * RD_STRIDE + NRES * TD_STRIDE) // 748 ints

#if defined(__HIP_DEVICE_COMPILE__) &&                                         \
    __has_builtin(__builtin_amdgcn_global_load_async_to_lds_b128) &&           \
    __has_builtin(__builtin_amdgcn_global_load_async_to_lds_b32)
#define HAVE_ASYNC_LDS 1
#else
#define HAVE_ASYNC_LDS 0
#endif

typedef int vint4 __attribute__((ext_vector_type(4)));
typedef __attribute__((address_space(1))) vint4 g_vint4;
typedef __attribute__((address_space(3))) vint4 l_vint4;
typedef __attribute__((address_space(1))) int   g_int;
typedef __attribute__((address_space(3))) int   l_int;

__device__ __forceinline__ void copy16_async(void* lds_dst, const void* gsrc) {
#if HAVE_ASYNC_LDS
  __builtin_amdgcn_global_load_async_to_lds_b128(
      (g_vint4*)(gsrc), (l_vint4*)(lds_dst), 0, 0);
#else
  *(float4*)lds_dst = *(const float4*)gsrc;
#endif
}

__device__ __forceinline__ void copy4_async(void* lds_dst, const void* gsrc) {
#if HAVE_ASYNC_LDS
  __builtin_amdgcn_global_load_async_to_lds_b32(
      (g_int*)(gsrc), (l_int*)(lds_dst), 0, 0);
#else
  *(unsigned*)lds_dst = *(const unsigned*)gsrc;
#endif
}

__device__ __forceinline__ void wait_async0() {
#if HAVE_ASYNC_LDS
#if __has_builtin(__builtin_amdgcn_s_wait_asynccnt)
  __builtin_amdgcn_s_wait_asynccnt(0);
#else
  asm volatile("s_wait_asynccnt 0" ::: "memory");
#endif
#endif
}

struct Frame { float r[9]; float t[3]; };

// Z = X o Y : Zr = Xr*Yr ; Zt = Xr*Yt + Xt   (rows of r are matrix rows)
__device__ __forceinline__ Frame combine(const Frame& X, const Frame& Y) {
  Frame Z;
#pragma unroll
  for (int i = 0; i < 3; ++i) {
#pragma unroll
    for (int j = 0; j < 3; ++j)
      Z.r[i*3+j] = X.r[i*3+0]*Y.r[0+j] + X.r[i*3+1]*Y.r[3+j] + X.r[i*3+2]*Y.r[6+j];
    Z.t[i] = X.r[i*3+0]*Y.t[0] + X.r[i*3+1]*Y.t[1] + X.r[i*3+2]*Y.t[2] + X.t[i];
  }
  return Z;
}

// select op[p] from compile-time-indexed register array (7 cmp + cndmask tree)
__device__ __forceinline__ Frame select8(const Frame op[8], int p) {
  Frame o = op[0];
#pragma unroll
  for (int q = 1; q < 8; ++q) {
    bool c = (p == q);
#pragma unroll
    for (int k = 0; k < 9; ++k) o.r[k] = c ? op[q].r[k] : o.r[k];
#pragma unroll
    for (int k = 0; k < 3; ++k) o.t[k] = c ? op[q].t[k] : o.t[k];
  }
  return o;
}

extern "C" __global__ __launch_bounds__(BLK)
void frames_atoms_kernel(const float* __restrict__ out0,   // (N,17)
                         const float* __restrict__ out1,   // (N,9)
                         const float* __restrict__ posv,   // (N,3)
                         const int*   __restrict__ ssv,    // (N)
                         const int*   __restrict__ rtv,    // (N)
                         const float* __restrict__ gT,     // (3,22,8,4,3) = 6336
                         const float* __restrict__ gG,     // (3,22,24,3)  = 4752
                         const int*   __restrict__ gTD,    // (22,8)  = 176
                         const int*   __restrict__ gRD,    // (22,24) = 528
                         float* __restrict__ oR,           // (N,24,3)
                         float* __restrict__ oBB,          // (N,4,3)
                         int n_total) {
  __shared__ __align__(16) float s_mem[A_DW + B_DW]; // 14616 dw = 58464 B
  __shared__ __align__(16) int   s_dep[DEP_N];       // 748 ints = 2992 B
  float* sA = s_mem;            // union: transforms table, later frame scratch
  float* sB = s_mem + A_DW;     // rigids table
  const int tid = threadIdx.x;

  // ---- stage tables into LDS (async DMA path on gfx1250) ----
  // transforms: 1584 x b128 chunks, 24 chunks per 96-dw entry, pad to 100 dw
#pragma unroll 1
  for (int c = tid; c < 1584; c += BLK) {
    int e = c / 24, o = (c % 24) * 4;
    copy16_async(sA + e * T_STRIDE + o, gT + c * 4);
  }
  // rigids: 1188 x b128 chunks, 18 per 72-dw entry, pad to 76 dw
#pragma unroll 1
  for (int c = tid; c < 1188; c += BLK) {
    int e = c / 18, o = (c % 18) * 4;
    copy16_async(sB + e * G_STRIDE + o, gG + c * 4);
  }
#pragma unroll 1
  for (int i = tid; i < NRES * MAX_ATOM; i += BLK)
    copy4_async(s_dep + DEP_RD + (i / MAX_ATOM) * RD_STRIDE + (i % MAX_ATOM), gRD + i);
#pragma unroll 1
  for (int i = tid; i < NRES * MAX_RIGID; i += BLK)
    copy4_async(s_dep + DEP_TD + (i / MAX_RIGID) * TD_STRIDE + (i % MAX_RIGID), gTD + i);
  wait_async0();
  __syncthreads();

  const int  n      = (int)blockIdx.x * BLK + tid;
  const bool active = (n < n_total);
  const int  nc     = active ? n : 0;

  const int si = ssv[nc];
  const int ri = rtv[nc];
  const int e  = si * NRES + ri;

  // ---- backbone frame: Gram-Schmidt on output_1 rows 0/1, cross for e2 ----
  const float* o1 = out1 + (size_t)nc * 9;
  float v0x = o1[0], v0y = o1[1], v0z = o1[2];
  float v1x = o1[3], v1y = o1[4], v1z = o1[5];
  float inv0 = 1.0f / fmaxf(sqrtf(v0x*v0x + v0y*v0y + v0z*v0z), EPSF);
  float e0x = v0x*inv0, e0y = v0y*inv0, e0z = v0z*inv0;
  float d   = e0x*v1x + e0y*v1y + e0z*v1z;
  float u1x = v1x - e0x*d, u1y = v1y - e0y*d, u1z = v1z - e0z*d;
  float inv1 = 1.0f / fmaxf(sqrtf(u1x*u1x + u1y*u1y + u1z*u1z), EPSF);
  float e1x = u1x*inv1, e1y = u1y*inv1, e1z = u1z*inv1;
  float e2x = e0y*e1z - e0z*e1y;
  float e2y = e0z*e1x - e0x*e1z;
  float e2z = e0x*e1y - e0y*e1x;

  Frame op[8];
  // rot columns are (e0,e1,e2)  -> row i = (e0[i], e1[i], e2[i])
  op[0].r[0] = e0x; op[0].r[1] = e1x; op[0].r[2] = e2x;
  op[0].r[3] = e0y; op[0].r[4] = e1y; op[0].r[5] = e2y;
  op[0].r[6] = e0z; op[0].r[7] = e1z; op[0].r[8] = e2z;
  const float* pp = posv + (size_t)nc * 3;
  op[0].t[0] = 0.1f * o1[6] + pp[0];
  op[0].t[1] = 0.1f * o1[7] + pp[1];
  op[0].t[2] = 0.1f * o1[8] + pp[2];

  // ---- 7 torsion X-rotations from normalized (cos,sin) ----
  const float* o0 = out0 + (size_t)nc * 17;
#pragma unroll
  for (int k = 0; k < MAX_TORSION; ++k) {
    float c = o0[2*k], s = o0[2*k + 1];
    float il = 1.0f / fmaxf(sqrtf(c*c + s*s), EPSF);
    c *= il; s *= il;
    Frame& f = op[k + 1];
    f.r[0] = 1.f; f.r[1] = 0.f; f.r[2] = 0.f;
    f.r[3] = 0.f; f.r[4] = c;   f.r[5] = -s;
    f.r[6] = 0.f; f.r[7] = s;   f.r[8] = c;
    f.t[0] = 0.f; f.t[1] = 0.f; f.t[2] = 0.f;
  }

  // ---- compose with table transforms (LDS gather, b128-merged) ----
  {
    const float* T = sA + (size_t)e * T_STRIDE;
#pragma unroll
    for (int g = 0; g < MAX_RIGID; ++g) {
      Frame Tg;
#pragma unroll
      for (int k = 0; k < 9; ++k) Tg.r[k] = T[g*12 + k];
#pragma unroll
      for (int k = 0; k < 3; ++k) Tg.t[k] = T[g*12 + 9 + k];
      op[g] = combine(Tg, op[g]);
    }
  }

  // ---- data-dependent composition chain (sequential, matches reference) ----
#pragma unroll
  for (int i = 1; i < MAX_RIGID; ++i) {
    int p = s_dep[DEP_TD + ri * TD_STRIDE + i];
    Frame prev = select8(op, p);
    op[i] = combine(prev, op[i]);
  }

  // ---- second output: opr[:,0] as (4,3) row-major ----
  if (active) {
    float* ob = oBB + (size_t)n * 12;
    *(float4*)(ob + 0) = make_float4(op[0].r[0], op[0].r[1], op[0].r[2], op[0].r[3]);
    *(float4*)(ob + 4) = make_float4(op[0].r[4], op[0].r[5], op[0].r[6], op[0].r[7]);
    *(float4*)(ob + 8) = make_float4(op[0].r[8], op[0].t[0], op[0].t[1], op[0].t[2]);
  }

  // ---- transforms region is dead: reuse it as per-thread frame scratch ----
  __syncthreads();  // all lanes done reading transforms from sA
  float* fb = sA + tid * FR_STRIDE;  // 400 B/thread, 16B aligned
#pragma unroll
  for (int g = 0; g < MAX_RIGID; ++g) {
#pragma unroll
    for (int k = 0; k < 9; ++k) fb[g*12 + k] = op[g].r[k];
    fb[g*12 +  9] = op[g].t[0];
    fb[g*12 + 10] = op[g].t[1];
    fb[g*12 + 11] = op[g].t[2];
  }
  // Only the owning thread reads its own scratch -> wave-local DS ordering
  // suffices (DScnt), no extra barrier needed.

  // ---- atoms: gather frame via rigids_dep (3x ds_load_b128), matvec ----
  const float* rg = sB + (size_t)e * G_STRIDE;
#pragma unroll
  for (int a0 = 0; a0 < MAX_ATOM; a0 += 4) {
    float buf[12];
#pragma unroll
    for (int u = 0; u < 4; ++u) {
      int a = a0 + u;
      int g = s_dep[DEP_RD + ri * RD_STRIDE + a];
      const float* F = fb + g * 12;
      float x = rg[a*3 + 0], y = rg[a*3 + 1], z = rg[a*3 + 2];
      buf[u*3 + 0] = F[0]*x + F[1]*y + F[2]*z + F[9];
      buf[u*3 + 1] = F[3]*x + F[4]*y + F[5]*z + F[10];
      buf[u*3 + 2] = F[6]*x + F[7]*y + F[8]*z + F[11];
    }
    if (active) {
      float* po = oR + (size_t)n * 72 + a0 * 3;   // 48B offset, 16B aligned
      *(float4*)(po + 0) = make_float4(buf[0], buf[1], buf[2],  buf[3]);
      *(float4*)(po + 4) = make_float4(buf[4], buf[5], buf[6],  buf[7]);
      *(float4*)(po + 8) = make_float4(buf[8], buf[9], buf[10], buf[11]);
    }
  }
}

extern "C" void kernel_launch(void* const* d_in, const int* in_sizes, int n_in,
                              void* d_out, int out_size, void* d_ws, size_t ws_size,
                              hipStream_t stream) {
  (void)n_in; (void)out_size; (void)d_ws; (void)ws_size;
  const float* out0 = (const float*)d_in[0];
  const float* out1 = (const float*)d_in[1];
  const float* posv = (const float*)d_in[2];
  const int*   ssv  = (const int*)d_in[3];
  const int*   rtv  = (const int*)d_in[4];
  const float* gT   = (const float*)d_in[5];
  const float* gG   = (const float*)d_in[6];
  const int*   gTD  = (const int*)d_in[7];
  const int*   gRD  = (const int*)d_in[8];
  const int n = in_sizes[3];            // ss has N elements
  float* oR  = (float*)d_out;           // (N,24,3) first
  float* oBB = oR + (size_t)n * 72;     // (N,4,3) second
  const int grid = (n + BLK - 1) / BLK; // 3125 for N=300000 (exact)
  frames_atoms_kernel<<<grid, BLK, 0, stream>>>(out0, out1, posv, ssv, rtv,
                                                gT, gG, gTD, gRD, oR, oBB, n);
}